// MyLayer_61289183314546
// MI455X (gfx1250) — compile-verified
//
#include <hip/hip_runtime.h>
#include <hip/hip_bf16.h>
#include <math.h>

#define N_NODES 20000
#define N_EDGES 256000
#define M_PATHS 3
#define IN_SIZE 256
#define HEADS 8
#define OUT_F 64
#define D_DIM 512
#define HID 128
#define NEG_SLOPE 0.2f

typedef float v2f __attribute__((ext_vector_type(2)));
typedef float v8f __attribute__((ext_vector_type(8)));

// ---- order-preserving float <-> uint encoding for atomic segment-max ----
__device__ __forceinline__ unsigned enc_f32(float x) {
    unsigned u = __float_as_uint(x);
    return (u & 0x80000000u) ? ~u : (u | 0x80000000u);
}
__device__ __forceinline__ float dec_f32(unsigned u) {
    unsigned b = (u & 0x80000000u) ? (u & 0x7FFFFFFFu) : ~u;
    return __uint_as_float(b);
}

__device__ __forceinline__ v8f wmma_f32(v2f a, v2f b, v8f c) {
    return __builtin_amdgcn_wmma_f32_16x16x4_f32(
        /*neg_a=*/false, a, /*neg_b=*/false, b,
        /*c_mod=*/(short)0, c, /*reuse_a=*/false, /*reuse_b=*/false);
}

// =====================================================================
// Kernel 1: feat = h @ W[p]   (M=20000, N=512, K=256) via f32 WMMA.
// Register-blocked: each wave computes a 32x64 output block
// (2 M-tiles x 4 N-tiles, 8 accumulators). Per K-step: 2 A-loads +
// 8 B-loads feed 8 WMMAs (1.25 loads/WMMA vs 3 unblocked).
// 625 m-groups x 8 n-groups = 5000 waves, exact tiling.
// =====================================================================
__global__ void __launch_bounds__(128)
gemm_feat_kernel(const float* __restrict__ h, const float* __restrict__ Wp,
                 float* __restrict__ feat) {
    const int lane = threadIdx.x & 31;
    const int wave = blockIdx.x * 4 + (threadIdx.x >> 5);
    const int ng   = wave & 7;    // n-group: 64 cols
    const int mg   = wave >> 3;   // m-group: 32 rows (0..624)
    const int half = lane >> 4;   // 0: K{0,1}, 1: K{2,3}
    const int r    = lane & 15;

    const int arow0 = mg * 32 + r;        // A rows, m-tile 0
    const int arow1 = arow0 + 16;         // A rows, m-tile 1
    const int bcol  = ng * 64 + r;        // B col base for this lane

    v8f c[2][4];
#pragma unroll
    for (int mi = 0; mi < 2; ++mi)
#pragma unroll
        for (int j = 0; j < 4; ++j) c[mi][j] = (v8f){};

    for (int k = 0; k < IN_SIZE; k += 4) {
        const int k0 = k + half * 2;
        v2f a0 = *(const v2f*)(h + (size_t)arow0 * IN_SIZE + k0);
        v2f a1 = *(const v2f*)(h + (size_t)arow1 * IN_SIZE + k0);
        const float* w0 = Wp + (size_t)k0 * D_DIM + bcol;  // row k0
        const float* w1 = w0 + D_DIM;                      // row k0+1
        v2f b[4];
#pragma unroll
        for (int j = 0; j < 4; ++j) {
            b[j].x = w0[16 * j];
            b[j].y = w1[16 * j];
        }
#pragma unroll
        for (int j = 0; j < 4; ++j) {
            c[0][j] = wmma_f32(a0, b[j], c[0][j]);
            c[1][j] = wmma_f32(a1, b[j], c[1][j]);
        }
    }

    // C/D layout: VGPR i -> M = half*8 + i; N = lane & 15
    const int crow = mg * 32 + half * 8;
    const int ccol = ng * 64 + r;
#pragma unroll
    for (int mi = 0; mi < 2; ++mi)
#pragma unroll
        for (int j = 0; j < 4; ++j)
#pragma unroll
            for (int i = 0; i < 8; ++i)
                feat[(size_t)(crow + mi * 16 + i) * D_DIM + ccol + 16 * j] =
                    c[mi][j][i];
}

// =====================================================================
// Kernel 2: el[n,h] = <feat[n,h,:], attn_l[h,:]>, er likewise
// =====================================================================
__global__ void __launch_bounds__(256)
elr_kernel(const float* __restrict__ feat, const float* __restrict__ al,
           const float* __restrict__ ar, float* __restrict__ el,
           float* __restrict__ er) {
    int idx = blockIdx.x * blockDim.x + threadIdx.x;
    if (idx >= N_NODES * HEADS) return;
    const int n  = idx >> 3;
    const int hd = idx & 7;
    const float* f  = feat + (size_t)n * D_DIM + hd * OUT_F;
    const float* pl = al + hd * OUT_F;
    const float* pr = ar + hd * OUT_F;
    float sl = 0.f, sr = 0.f;
#pragma unroll 4
    for (int i = 0; i < OUT_F; ++i) {
        float v = f[i];
        sl += v * pl[i];
        sr += v * pr[i];
    }
    el[idx] = sl;
    er[idx] = sr;
}

// =====================================================================
// Kernel 3: init per-path segment-max / denom buffers
// =====================================================================
__global__ void __launch_bounds__(256)
init_mdenom_kernel(unsigned* __restrict__ m_enc, float* __restrict__ denom) {
    int idx = blockIdx.x * blockDim.x + threadIdx.x;
    if (idx >= N_NODES * HEADS) return;
    m_enc[idx] = enc_f32(-INFINITY);
    denom[idx] = 0.f;
}

// =====================================================================
// Kernel 4: e = leaky_relu(el[src]+er[dst]); segment max via atomicMax
// =====================================================================
__global__ void __launch_bounds__(256)
edge_e_kernel(const int* __restrict__ src, const int* __restrict__ dst,
              const float* __restrict__ el, const float* __restrict__ er,
              float* __restrict__ ebuf, unsigned* __restrict__ m_enc) {
    int idx = blockIdx.x * blockDim.x + threadIdx.x;
    if (idx >= N_EDGES * HEADS) return;
    const int e  = idx >> 3;
    const int hd = idx & 7;
    const int s = src[e];
    const int d = dst[e];
    float v = el[s * HEADS + hd] + er[d * HEADS + hd];
    v = (v > 0.f) ? v : NEG_SLOPE * v;
    ebuf[idx] = v;
    atomicMax(m_enc + d * HEADS + hd, enc_f32(v));
}

// =====================================================================
// Kernel 5: ex = exp(e - m[dst]); denom[dst] += ex (segment sum)
// =====================================================================
__global__ void __launch_bounds__(256)
edge_ex_kernel(const int* __restrict__ dst, float* __restrict__ ebuf,
               const unsigned* __restrict__ m_enc, float* __restrict__ denom) {
    int idx = blockIdx.x * blockDim.x + threadIdx.x;
    if (idx >= N_EDGES * HEADS) return;
    const int e  = idx >> 3;
    const int hd = idx & 7;
    const int d  = dst[e];
    float m  = dec_f32(m_enc[d * HEADS + hd]);
    float ex = expf(ebuf[idx] - m);
    ebuf[idx] = ex;
    atomicAdd(denom + d * HEADS + hd, ex);
}

// =====================================================================
// Kernel 6: alpha = ex/(denom+1e-9); write alpha out; z[dst,p,:] += alpha*feat[src,:]
// =====================================================================
__global__ void __launch_bounds__(256)
edge_scatter_kernel(const int* __restrict__ src, const int* __restrict__ dst,
                    const float* __restrict__ ebuf, const float* __restrict__ denom,
                    const float* __restrict__ feat, float* __restrict__ alpha_out,
                    float* __restrict__ z, int path) {
    int idx = blockIdx.x * blockDim.x + threadIdx.x;
    if (idx >= N_EDGES * HEADS) return;
    const int e  = idx >> 3;
    const int hd = idx & 7;
    const int s = src[e];
    const int d = dst[e];
    const float alpha = ebuf[idx] / (denom[d * HEADS + hd] + 1e-9f);
    alpha_out[(size_t)e * HEADS + hd] = alpha;

    const float4* fs = (const float4*)(feat + (size_t)s * D_DIM + hd * OUT_F);
    float* zd = z + ((size_t)d * M_PATHS + path) * D_DIM + hd * OUT_F;
#pragma unroll 4
    for (int q = 0; q < OUT_F / 4; ++q) {
        float4 v = fs[q];
        atomicAdd(zd + 4 * q + 0, alpha * v.x);
        atomicAdd(zd + 4 * q + 1, alpha * v.y);
        atomicAdd(zd + 4 * q + 2, alpha * v.z);
        atomicAdd(zd + 4 * q + 3, alpha * v.w);
    }
}

// =====================================================================
// Kernel 7: z = elu(z + bias[m])  over [N, M, D]
// =====================================================================
__global__ void __launch_bounds__(256)
finalize_z_kernel(float* __restrict__ z, const float* __restrict__ bias) {
    long idx = (long)blockIdx.x * blockDim.x + threadIdx.x;
    if (idx >= (long)N_NODES * M_PATHS * D_DIM) return;
    const int m  = (int)((idx / D_DIM) % M_PATHS);
    const int dd = (int)(idx % D_DIM);
    float v = z[idx] + bias[m * D_DIM + dd];
    z[idx] = (v > 0.f) ? v : (expf(v) - 1.f);
}

// =====================================================================
// Kernel 8: semantic attention logits; wave per (n,m); 4 hidden units/lane
// =====================================================================
__global__ void __launch_bounds__(256)
semantic_kernel(const float* __restrict__ z, const float* __restrict__ w1,
                const float* __restrict__ b1, const float* __restrict__ w2,
                float* __restrict__ wsum) {
    const int wave = (int)((blockIdx.x * blockDim.x + threadIdx.x) >> 5);
    const int lane = threadIdx.x & 31;
    if (wave >= N_NODES * M_PATHS) return;
    const float* zr = z + (size_t)wave * D_DIM;  // wave == n*M + m matches z layout
    float s0 = b1[lane], s1 = b1[lane + 32], s2 = b1[lane + 64], s3 = b1[lane + 96];
    for (int f = 0; f < D_DIM; ++f) {
        const float zv = zr[f];
        const float* wr = w1 + (size_t)f * HID + lane;
        s0 += zv * wr[0];
        s1 += zv * wr[32];
        s2 += zv * wr[64];
        s3 += zv * wr[96];
    }
    float acc = tanhf(s0) * w2[lane] + tanhf(s1) * w2[lane + 32] +
                tanhf(s2) * w2[lane + 64] + tanhf(s3) * w2[lane + 96];
    for (int off = 16; off > 0; off >>= 1) acc += __shfl_down(acc, off, 32);
    if (lane == 0) atomicAdd(wsum + (wave % M_PATHS), acc);
}

// =====================================================================
// Kernel 9: beta = softmax(mean_n w)  (single thread, M=3)
// =====================================================================
__global__ void beta_kernel(const float* __restrict__ wsum,
                            float* __restrict__ beta_ws,
                            float* __restrict__ beta_out) {
    float w0 = wsum[0] / (float)N_NODES;
    float w1 = wsum[1] / (float)N_NODES;
    float w2 = wsum[2] / (float)N_NODES;
    float mx = fmaxf(w0, fmaxf(w1, w2));
    float e0 = expf(w0 - mx), e1 = expf(w1 - mx), e2 = expf(w2 - mx);
    float s = e0 + e1 + e2;
    beta_ws[0] = e0 / s; beta_ws[1] = e1 / s; beta_ws[2] = e2 / s;
    beta_out[0] = e0 / s; beta_out[1] = e1 / s; beta_out[2] = e2 / s;
}

// =====================================================================
// Kernel 10: gat_out[n,d] = sum_m beta[m] * z[n,m,d]
// =====================================================================
__global__ void __launch_bounds__(256)
combine_kernel(const float* __restrict__ z, const float* __restrict__ beta,
               float* __restrict__ out) {
    long idx = (long)blockIdx.x * blockDim.x + threadIdx.x;
    if (idx >= (long)N_NODES * D_DIM) return;
    const int n  = (int)(idx / D_DIM);
    const int dd = (int)(idx % D_DIM);
    const float* zr = z + (size_t)n * M_PATHS * D_DIM + dd;
    out[idx] = beta[0] * zr[0] + beta[1] * zr[D_DIM] + beta[2] * zr[2 * D_DIM];
}

// zero-fill helper (graph-capture safe)
__global__ void __launch_bounds__(256)
zero_kernel(float* __restrict__ p, long n) {
    long idx = (long)blockIdx.x * blockDim.x + threadIdx.x;
    if (idx < n) p[idx] = 0.f;
}

extern "C" void kernel_launch(void* const* d_in, const int* in_sizes, int n_in,
                              void* d_out, int out_size, void* d_ws, size_t ws_size,
                              hipStream_t stream) {
    const float* h      = (const float*)d_in[0];
    const float* W      = (const float*)d_in[1];
    const float* attn_l = (const float*)d_in[2];
    const float* attn_r = (const float*)d_in[3];
    const float* bias   = (const float*)d_in[4];
    const float* sem_w1 = (const float*)d_in[5];
    const float* sem_b1 = (const float*)d_in[6];
    const float* sem_w2 = (const float*)d_in[7];
    const int*   src    = (const int*)d_in[8];
    const int*   dst    = (const int*)d_in[9];
    float* out = (float*)d_out;

    // workspace carve-up (floats)
    float* ws = (float*)d_ws;
    size_t off = 0;
    float*    feat  = ws + off; off += (size_t)N_NODES * D_DIM;             // 10.24M
    float*    z     = ws + off; off += (size_t)N_NODES * M_PATHS * D_DIM;   // 30.72M
    float*    el    = ws + off; off += (size_t)N_NODES * HEADS;
    float*    er    = ws + off; off += (size_t)N_NODES * HEADS;
    unsigned* m_enc = (unsigned*)(ws + off); off += (size_t)N_NODES * HEADS;
    float*    denom = ws + off; off += (size_t)N_NODES * HEADS;
    float*    ebuf  = ws + off; off += (size_t)N_EDGES * HEADS;
    float*    wsum  = ws + off; off += 4;
    float*    betaw = ws + off; off += 4;

    const long zN = (long)N_NODES * M_PATHS * D_DIM;
    zero_kernel<<<(int)((zN + 255) / 256), 256, 0, stream>>>(z, zN);
    zero_kernel<<<1, 256, 0, stream>>>(wsum, 4);

    const int nhB = (N_NODES * HEADS + 255) / 256;
    const int ehB = (N_EDGES * HEADS + 255) / 256;
    float* alpha_base = out + (size_t)N_NODES * D_DIM;

    for (int p = 0; p < M_PATHS; ++p) {
        const float* Wp  = W + (size_t)p * IN_SIZE * D_DIM;
        const float* alp = attn_l + (size_t)p * HEADS * OUT_F;
        const float* arp = attn_r + (size_t)p * HEADS * OUT_F;
        const int* sp = src + (size_t)p * N_EDGES;
        const int* dp = dst + (size_t)p * N_EDGES;

        // 625 m-groups * 8 n-groups = 5000 waves, 4 waves/block
        gemm_feat_kernel<<<1250, 128, 0, stream>>>(h, Wp, feat);
        elr_kernel<<<nhB, 256, 0, stream>>>(feat, alp, arp, el, er);
        init_mdenom_kernel<<<nhB, 256, 0, stream>>>(m_enc, denom);
        edge_e_kernel<<<ehB, 256, 0, stream>>>(sp, dp, el, er, ebuf, m_enc);
        edge_ex_kernel<<<ehB, 256, 0, stream>>>(dp, ebuf, m_enc, denom);
        edge_scatter_kernel<<<ehB, 256, 0, stream>>>(
            sp, dp, ebuf, denom, feat,
            alpha_base + (size_t)p * N_EDGES * HEADS, z, p);
    }

    finalize_z_kernel<<<(int)((zN + 255) / 256), 256, 0, stream>>>(z, bias);
    // 60000 waves, 8 waves/block -> 7500 blocks
    semantic_kernel<<<7500, 256, 0, stream>>>(z, sem_w1, sem_b1, sem_w2, wsum);
    beta_kernel<<<1, 1, 0, stream>>>(
        wsum, betaw, out + (size_t)N_NODES * D_DIM + (size_t)M_PATHS * N_EDGES * HEADS);
    const long oN = (long)N_NODES * D_DIM;
    combine_kernel<<<(int)((oN + 255) / 256), 256, 0, stream>>>(z, betaw, out);
}